// SdfParseLoss_23708219474147
// MI455X (gfx1250) — compile-verified
//
#include <hip/hip_runtime.h>
#include <hip/hip_bf16.h>

// ---------------- problem constants (match reference) ----------------
#define RH 256
#define RW 192
#define RHW (RH * RW)          // 49152
#define RB 64
#define RV 100000
#define RNB 48                 // partial blocks per batch (1024 pixels/block)
#define R_INF 9999.0f
#define R_INIT 1.0e30f

typedef float v2f __attribute__((ext_vector_type(2)));
typedef float v8f __attribute__((ext_vector_type(8)));

// Order-preserving float <-> uint32 key mapping: unsigned compare on keys
// == float compare on values. Guarantees single-instruction
// GLOBAL_ATOMIC_MIN_U32 / MAX_U32 (no FP-atomic CAS-loop risk).
__device__ __forceinline__ unsigned int f2key(float f) {
    unsigned int u = __float_as_uint(f);
    return (u & 0x80000000u) ? ~u : (u | 0x80000000u);
}
__device__ __forceinline__ float key2f(unsigned int k) {
    unsigned int u = (k & 0x80000000u) ? (k ^ 0x80000000u) : ~k;
    return __uint_as_float(u);
}

__device__ __forceinline__ float wave_sum(float v) {
    for (int off = 16; off > 0; off >>= 1)
        v += __shfl_down(v, off, 32);
    return v;
}

// ---------------- kernel 1: init key sentinels ----------------
__global__ void k_init(uint4* __restrict__ kmin4, uint4* __restrict__ kmax4, int n4) {
    int i = blockIdx.x * blockDim.x + threadIdx.x;
    if (i < n4) {
        const unsigned int kmn = f2key(R_INIT);    // +1e30 key (folds to const)
        const unsigned int kmx = f2key(-R_INIT);   // -1e30 key
        kmin4[i] = make_uint4(kmn, kmn, kmn, kmn);
        kmax4[i] = make_uint4(kmx, kmx, kmx, kmx);
    }
}

// ---------------- kernel 2: scatter min/max via u32 atomics on keys ----------------
__global__ void k_scatter(const float* __restrict__ sdf,
                          const float2* __restrict__ uv,
                          unsigned int* __restrict__ kmin,
                          unsigned int* __restrict__ kmax) {
    int v = blockIdx.x * blockDim.x + threadIdx.x;
    int b = blockIdx.y;
    if (v >= RV) return;
    int iv = b * RV + v;
    float2 c = uv[iv];
    int x = (int)c.x;            // trunc-toward-zero == jnp astype(int32)
    int y = (int)c.y;
    if (x >= 0 && x < RW && y >= 0 && y < RH) {
        unsigned int k = f2key(sdf[iv]);
        int p = b * RHW + y * RW + x;
        atomicMin(&kmin[p], k);  // global_atomic_min_u32
        atomicMax(&kmax[p], k);  // global_atomic_max_u32
    }
}

// ---------------- kernel 3: per-(b,block) masked partial sums ----------------
// 256 threads x 4 pixels (uint4/int4 loads) = 1024 pixels per block.
__global__ void k_partials(const uint4* __restrict__ kmin4,
                           const uint4* __restrict__ kmax4,
                           const int4* __restrict__ gt4,
                           const float* __restrict__ thresh_p,
                           const float* __restrict__ pvalid,
                           float* __restrict__ ppos,
                           float* __restrict__ pneg,
                           float* __restrict__ pcnt) {
    const int tid = threadIdx.x;
    const int b = blockIdx.y;
    const float th = thresh_p[0];
    const float pv = pvalid[b];
    const int i4 = b * (RHW / 4) + blockIdx.x * 256 + tid;

    uint4 km = kmin4[i4];
    uint4 kx = kmax4[i4];
    int4  g  = gt4[i4];

    float lp = 0.0f, ln = 0.0f, cnt = 0.0f;
    {
        unsigned int kmv[4] = {km.x, km.y, km.z, km.w};
        unsigned int kxv[4] = {kx.x, kx.y, kx.z, kx.w};
        int gv[4] = {g.x, g.y, g.z, g.w};
        #pragma unroll
        for (int e = 0; e < 4; ++e) {
            float mn = key2f(kmv[e]);
            float mx = key2f(kxv[e]);
            mn = (mn >= R_INF) ? 0.0f : mn;
            mx = (mx <= -R_INF) ? th : mx;
            float pos = (gv[e] == 1) ? 1.0f : 0.0f;
            float neg = (gv[e] == 0) ? 1.0f : 0.0f;
            lp += fabsf(mn) * pos;
            ln += fabsf(mx - th) * neg;
            cnt += pos;
        }
    }
    lp *= pv;
    ln *= pv;

    // wave32 shuffle reduction, then cross-wave LDS
    lp = wave_sum(lp);
    ln = wave_sum(ln);
    cnt = wave_sum(cnt);

    __shared__ float sp[8], sn[8], sc[8];
    const int wid = tid >> 5, lid = tid & 31;
    if (lid == 0) { sp[wid] = lp; sn[wid] = ln; sc[wid] = cnt; }
    __syncthreads();
    if (tid == 0) {
        float a = 0.0f, bb = 0.0f, cc = 0.0f;
        #pragma unroll
        for (int w = 0; w < 8; ++w) { a += sp[w]; bb += sn[w]; cc += sc[w]; }
        int o = b * RNB + blockIdx.x;
        ppos[o] = a;
        pneg[o] = bb;
        pcnt[o] = cc;
    }
}

// ---------------- kernel 4: final reduction as GEMV via WMMA ----------------
// Sums partials[64][48] * ones[48] -> [64] with V_WMMA_F32_16X16X4_F32.
// A = 16x4 partial tiles, B = all-ones 4x16, D accumulates row sums
// (replicated across columns). One wave; EXEC all-1s during WMMA loop.
__global__ void k_final(const float* __restrict__ ppos,
                        const float* __restrict__ pneg,
                        const float* __restrict__ pcnt,
                        float* __restrict__ out) {
    const int lane = threadIdx.x;          // 0..31
    const int m = lane & 15;               // A row within tile
    const int kofs = (lane >> 4) << 1;     // lanes 0-15: K0/K1, lanes 16-31: K2/K3
    const float invHW = 1.0f / (float)RHW;

#if __has_builtin(__builtin_amdgcn_wmma_f32_16x16x4_f32)
    v2f ones; ones.x = 1.0f; ones.y = 1.0f;
    for (int tm = 0; tm < 4; ++tm) {
        v8f dp = {}; v8f dn = {}; v8f dc = {};
        const int row = tm * 16 + m;
        const float* rp = ppos + row * RNB;
        const float* rn = pneg + row * RNB;
        const float* rc = pcnt + row * RNB;
        for (int k0 = 0; k0 < RNB; k0 += 4) {
            v2f a;
            a.x = rp[k0 + kofs]; a.y = rp[k0 + kofs + 1];
            dp = __builtin_amdgcn_wmma_f32_16x16x4_f32(false, a, false, ones,
                                                       (short)0, dp, false, false);
            a.x = rn[k0 + kofs]; a.y = rn[k0 + kofs + 1];
            dn = __builtin_amdgcn_wmma_f32_16x16x4_f32(false, a, false, ones,
                                                       (short)0, dn, false, false);
            a.x = rc[k0 + kofs]; a.y = rc[k0 + kofs + 1];
            dc = __builtin_amdgcn_wmma_f32_16x16x4_f32(false, a, false, ones,
                                                       (short)0, dc, false, false);
        }
        // D layout: lane 0 -> D[0..7][0] in dp[0..7]; lane 16 -> D[8..15][0]
        if (lane == 0) {
            for (int r = 0; r < 8; ++r) {
                float loss = (dp[r] + dn[r]) * invHW;
                out[tm * 16 + r] = (dc[r] > 0.0f) ? loss : 0.0f;
            }
        } else if (lane == 16) {
            for (int r = 0; r < 8; ++r) {
                float loss = (dp[r] + dn[r]) * invHW;
                out[tm * 16 + 8 + r] = (dc[r] > 0.0f) ? loss : 0.0f;
            }
        }
    }
#else
    // scalar fallback: each lane reduces 2 rows
    for (int rr = 0; rr < 2; ++rr) {
        int row = lane * 2 + rr;
        float sp = 0.0f, sn = 0.0f, sc = 0.0f;
        for (int k = 0; k < RNB; ++k) {
            sp += ppos[row * RNB + k];
            sn += pneg[row * RNB + k];
            sc += pcnt[row * RNB + k];
        }
        float loss = (sp + sn) * invHW;
        out[row] = (sc > 0.0f) ? loss : 0.0f;
    }
    (void)m; (void)kofs;
#endif
}

// ---------------- launcher ----------------
extern "C" void kernel_launch(void* const* d_in, const int* in_sizes, int n_in,
                              void* d_out, int out_size, void* d_ws, size_t ws_size,
                              hipStream_t stream) {
    const float*  sdf    = (const float*)d_in[0];   // [B,V]
    const float2* uv     = (const float2*)d_in[1];  // [B,V,2]
    const int*    gt     = (const int*)d_in[2];     // [B,H,W]
    const float*  thresh = (const float*)d_in[3];   // [1]
    const float*  pvalid = (const float*)d_in[5];   // [B,1,1]
    float* out = (float*)d_out;                     // [B]

    unsigned int* kmin = (unsigned int*)d_ws;       // B*HW keys
    unsigned int* kmax = kmin + (size_t)RB * RHW;   // B*HW keys
    float* ppos = (float*)(kmax + (size_t)RB * RHW);// B*NB
    float* pneg = ppos + (size_t)RB * RNB;          // B*NB
    float* pcnt = pneg + (size_t)RB * RNB;          // B*NB

    // 1) init key sentinels (vectorized)
    {
        int n4 = (RB * RHW) / 4;
        int blocks = (n4 + 255) / 256;
        k_init<<<blocks, 256, 0, stream>>>((uint4*)kmin, (uint4*)kmax, n4);
    }
    // 2) scatter min/max (u32 key atomics)
    {
        dim3 grid((RV + 255) / 256, RB);
        k_scatter<<<grid, 256, 0, stream>>>(sdf, uv, kmin, kmax);
    }
    // 3) masked per-block partial sums (vectorized loads + wave reduce)
    {
        dim3 grid(RNB, RB);
        k_partials<<<grid, 256, 0, stream>>>((const uint4*)kmin, (const uint4*)kmax,
                                             (const int4*)gt, thresh, pvalid,
                                             ppos, pneg, pcnt);
    }
    // 4) WMMA final reduction -> loss[B]
    {
        k_final<<<1, 32, 0, stream>>>(ppos, pneg, pcnt, out);
    }
    (void)in_sizes; (void)n_in; (void)out_size; (void)ws_size;
}